// LinearPredictiveCoefficientsStabilityCheck_55009941127440
// MI455X (gfx1250) — compile-verified
//
#include <hip/hip_runtime.h>
#include <stdint.h>

// LPC stability check: per-row Levinson step-down -> clip reflection coeffs -> step-up.
// B = 2,000,000 rows x 25 f32. One thread per row; LDS staging with CDNA5 async
// global<->LDS data movement (ASYNCcnt path).

#define ROWS_PER_BLOCK 256
#define MORD 24                         // number of reflection coefficients
#define ROW_LEN 25                      // gain + 24 coefficients
#define ELEMS_PER_BLOCK (ROWS_PER_BLOCK * ROW_LEN)   // 6400 floats = 25600 B LDS

__global__ __launch_bounds__(ROWS_PER_BLOCK)
void lpc_stab_kernel(const float* __restrict__ a, float* __restrict__ out, int nRows) {
    __shared__ float smem[ELEMS_PER_BLOCK];

    const int t     = threadIdx.x;
    const int base  = (int)blockIdx.x * ELEMS_PER_BLOCK;
    const int total = nRows * ROW_LEN;
    const bool full = (base + ELEMS_PER_BLOCK) <= total;   // uniform across block

    // ---------------- Stage global -> LDS (coalesced) ----------------
    if (full) {
        // CDNA5 async data mover: lane-coalesced b32 transfers, no VGPR data.
        const float* gp    = a + base + t;
        unsigned     lbase = (unsigned)(uintptr_t)(&smem[t]);  // low 32 bits = LDS byte offset
#pragma unroll
        for (int i = 0; i < ROW_LEN; ++i) {
            asm volatile("global_load_async_to_lds_b32 %0, %1, off"
                         :: "v"(lbase + (unsigned)(i * ROWS_PER_BLOCK * 4)),
                            "v"(gp + i * ROWS_PER_BLOCK)
                         : "memory");
        }
        asm volatile("s_wait_asynccnt 0x0" ::: "memory");
    } else {
        // tail block: guarded plain loads (async ops must run with EXEC all ones)
#pragma unroll
        for (int i = 0; i < ROW_LEN; ++i) {
            int g = base + t + i * ROWS_PER_BLOCK;
            smem[t + i * ROWS_PER_BLOCK] = (g < total) ? a[g] : 0.0f;
        }
    }
    __syncthreads();

    // ---------------- Per-row recursion, fully in registers ----------------
    // stride-25 word reads across lanes: gcd(25,64)==1 -> bank-conflict free
    float K = smem[t * ROW_LEN];
    float c[MORD];
#pragma unroll
    for (int j = 0; j < MORD; ++j) c[j] = smem[t * ROW_LEN + 1 + j];

    // step-down: c_new[j] = (c[j] - k*c[m-2-j]) / (1 - k^2), k_m = c[m-1]
    float k[MORD];
#pragma unroll
    for (int m = MORD; m >= 1; --m) {
        float km = c[m - 1];
        k[m - 1] = km;
        float inv = __builtin_amdgcn_rcpf(1.0f - km * km);
        int lo = 0, hi = m - 2;
#pragma unroll
        for (; lo < hi; ++lo, --hi) {       // pairwise: both use OLD values
            float cl = c[lo], ch = c[hi];
            c[lo] = (cl - km * ch) * inv;
            c[hi] = (ch - km * cl) * inv;
        }
        if (lo == hi) c[lo] = (c[lo] - km * c[lo]) * inv;   // middle element
    }

    // clip: float32(1 - 1e-16) == 1.0f, so the effective bound is +/-1.0f
#pragma unroll
    for (int j = 0; j < MORD; ++j) k[j] = fminf(fmaxf(k[j], -1.0f), 1.0f);

    // step-up: d_new[j] = d[j] + k*d[m-2-j]; d[m-1] = k_m
    float d[MORD];
#pragma unroll
    for (int j = 0; j < MORD; ++j) d[j] = 0.0f;
#pragma unroll
    for (int m = 1; m <= MORD; ++m) {
        float km = k[m - 1];
        int lo = 0, hi = m - 2;
#pragma unroll
        for (; lo < hi; ++lo, --hi) {
            float dl = d[lo], dh = d[hi];
            d[lo] = dl + km * dh;
            d[hi] = dh + km * dl;
        }
        if (lo == hi) d[lo] = d[lo] + km * d[lo];
        d[m - 1] = km;
    }

    // write result row back into the same LDS slab (own region only -> no barrier needed yet)
    smem[t * ROW_LEN] = K;
#pragma unroll
    for (int j = 0; j < MORD; ++j) smem[t * ROW_LEN + 1 + j] = d[j];
    __syncthreads();

    // ---------------- Drain LDS -> global (coalesced) ----------------
    if (full) {
        float*   op    = out + base + t;
        unsigned lbase = (unsigned)(uintptr_t)(&smem[t]);
#pragma unroll
        for (int i = 0; i < ROW_LEN; ++i) {
            asm volatile("global_store_async_from_lds_b32 %0, %1, off"
                         :: "v"(op + i * ROWS_PER_BLOCK),
                            "v"(lbase + (unsigned)(i * ROWS_PER_BLOCK * 4))
                         : "memory");
        }
        asm volatile("s_wait_asynccnt 0x0" ::: "memory");
    } else {
#pragma unroll
        for (int i = 0; i < ROW_LEN; ++i) {
            int g = base + t + i * ROWS_PER_BLOCK;
            if (g < total) out[g] = smem[t + i * ROWS_PER_BLOCK];
        }
    }
}

extern "C" void kernel_launch(void* const* d_in, const int* in_sizes, int n_in,
                              void* d_out, int out_size, void* d_ws, size_t ws_size,
                              hipStream_t stream) {
    const float* a   = (const float*)d_in[0];
    float*       out = (float*)d_out;

    const int nRows  = in_sizes[0] / ROW_LEN;           // 2,000,000
    const int blocks = (nRows + ROWS_PER_BLOCK - 1) / ROWS_PER_BLOCK;

    lpc_stab_kernel<<<blocks, ROWS_PER_BLOCK, 0, stream>>>(a, out, nRows);
}